// RDPP_NOISING_61692910240215
// MI455X (gfx1250) — compile-verified
//
#include <hip/hip_runtime.h>
#include <hip/hip_bf16.h>

// ---------------------------------------------------------------------------
// Problem constants (from the reference)
// ---------------------------------------------------------------------------
#define B_DIM   8
#define C_DIM   1024          // feature channels = GEMM K
#define HW_DIM  1024          // H*W = 32*32
#define N_ROWS  8192          // B*H*W query positions
#define M_DIM   16384         // memory bank entries
#define EPS_F   1e-8f
#define NOISE_MIN_F 0.01f
#define NOISE_RANGE_F 0.49f   // NOISE_MAX - NOISE_MIN

// LDS strides (floats), padded to dodge bank conflicts while keeping
// 8B/16B alignment for float2/float4 LDS accesses.
#define ASTR 1026             // A panel row stride: [16][ASTR]
#define BSTR 68               // B tile row stride:  [128][BSTR] (x2 buffers)

typedef __attribute__((ext_vector_type(2))) float v2f;
typedef __attribute__((ext_vector_type(8))) float v8f;
typedef int v4i_gcc __attribute__((vector_size(4 * sizeof(int))));

#define AS1 __attribute__((address_space(1)))
#define AS3 __attribute__((address_space(3)))

// ---------------------------------------------------------------------------
// CDNA5 async memory->LDS DMA (ASYNCcnt-tracked), builtin if available,
// inline-asm fallback otherwise.
// Probe-confirmed signature (round-2/3 diagnostics):
//   __builtin_amdgcn_global_load_async_to_lds_b128(v4i AS1*, v4i AS3*, Ii, Ii)
//   __builtin_amdgcn_global_load_async_to_lds_b32 (int AS1*, int AS3*, Ii, Ii)
// ---------------------------------------------------------------------------
#if defined(__has_builtin)
#if __has_builtin(__builtin_amdgcn_global_load_async_to_lds_b128)
#define HAVE_ASYNC_BUILTIN 1
#endif
#if __has_builtin(__builtin_amdgcn_s_wait_asynccnt)
#define HAVE_WAITASYNC_BUILTIN 1
#endif
#endif

__device__ __forceinline__ void async_ld_b128(const float* gp, float* lp) {
#ifdef HAVE_ASYNC_BUILTIN
    __builtin_amdgcn_global_load_async_to_lds_b128((AS1 v4i_gcc*)gp, (AS3 v4i_gcc*)lp, 0, 0);
#else
    unsigned loff = (unsigned)(unsigned long long)(AS3 void*)lp;
    asm volatile("global_load_async_to_lds_b128 %0, %1, off"
                 :: "v"(loff), "v"((unsigned long long)gp) : "memory");
#endif
}

__device__ __forceinline__ void async_ld_b32(const float* gp, float* lp) {
#ifdef HAVE_ASYNC_BUILTIN
    __builtin_amdgcn_global_load_async_to_lds_b32((AS1 int*)gp, (AS3 int*)lp, 0, 0);
#else
    unsigned loff = (unsigned)(unsigned long long)(AS3 void*)lp;
    asm volatile("global_load_async_to_lds_b32 %0, %1, off"
                 :: "v"(loff), "v"((unsigned long long)gp) : "memory");
#endif
}

__device__ __forceinline__ void wait_async_le8() {
#ifdef HAVE_WAITASYNC_BUILTIN
    __builtin_amdgcn_s_wait_asynccnt(8);
#else
    asm volatile("s_wait_asynccnt 0x8" ::: "memory");
#endif
}

__device__ __forceinline__ void wait_async_0() {
#ifdef HAVE_WAITASYNC_BUILTIN
    __builtin_amdgcn_s_wait_asynccnt(0);
#else
    asm volatile("s_wait_asynccnt 0x0" ::: "memory");
#endif
}

// ---------------------------------------------------------------------------
// Kernel 1: squared norms of feature rows and memory-bank rows.
// One wave per row (wave32). features layout is (B,C,H,W): f[n][c] lives at
// b*C*HW + c*HW + p with n = b*HW + p.
// ---------------------------------------------------------------------------
__global__ __launch_bounds__(256)
void norms_kernel(const float* __restrict__ features,
                  const float* __restrict__ mb,
                  float* __restrict__ fsqArr,
                  float* __restrict__ msqArr)
{
    const int wave = threadIdx.x >> 5;
    const int lane = threadIdx.x & 31;
    const int row  = blockIdx.x * 8 + wave;
    float s = 0.0f;
    if (row < N_ROWS) {
        const int b = row >> 10, p = row & 1023;
        const float* base = features + (size_t)b * (C_DIM * HW_DIM) + p;
        for (int c = lane; c < C_DIM; c += 32) {
            float v = base[(size_t)c * HW_DIM];
            s += v * v;
        }
    } else {
        const int m = row - N_ROWS;
        const float* base = mb + (size_t)m * C_DIM;
        for (int c = lane; c < C_DIM; c += 32) {
            float v = base[c];
            s += v * v;
        }
    }
#pragma unroll
    for (int off = 16; off > 0; off >>= 1) s += __shfl_xor(s, off, 32);
    if (lane == 0) {
        if (row < N_ROWS) fsqArr[row] = s;
        else              msqArr[row - N_ROWS] = s;
    }
}

// ---------------------------------------------------------------------------
// Kernel 2: fused fp32 WMMA GEMM (F @ MB^T) + per-row nearest-neighbor argmin.
// Grid: 512 WGs x 256 threads (8 waves, wave32). Each WG owns 16 query rows;
// its A panel (16 x 1024 fp32, 64KB) stays resident in LDS for the entire
// kernel (loaded once via async b32 DMA). The 8 waves split each 128-column
// tile (16 cols each); B chunks [128 x 64] are double-buffered in LDS and
// filled with GLOBAL_LOAD_ASYNC_TO_LDS_B128 DMAs so the next chunk streams
// while the current one feeds 16 WMMAs per wave. ASYNCcnt <= 8 guarantees
// the chunk being consumed has landed while the newest chunk is in flight.
//
// A-frag (16x4 f32): lanes 0-15 hold M=lane, K={k,k+1}; lanes 16-31 K={k+2,k+3}
// B-frag (4x16 f32): lanes 0-15 hold N=lane, K={k,k+1}; lanes 16-31 K={k+2,k+3}
// C/D (16x16 f32):   VGPR j -> M=j (lanes 0-15) / M=j+8 (lanes 16-31), N=lane%16
// ---------------------------------------------------------------------------
__global__ __launch_bounds__(256)
void knn_gemm_kernel(const float* __restrict__ features,
                     const float* __restrict__ mb,
                     const float* __restrict__ fsqArr,
                     const float* __restrict__ msqArr,
                     float* __restrict__ outDist,
                     int*   __restrict__ outIdx)
{
    extern __shared__ float smem[];
    float* sA   = smem;                         // [16][ASTR]
    float* sB   = smem + 16 * ASTR;             // 2 x [128][BSTR]
    float* redD = sB + 2 * 128 * BSTR;          // [128]
    int*   redI = (int*)(redD + 128);           // [128]

    const int tid     = threadIdx.x;
    const int wave    = tid >> 5;
    const int lane    = tid & 31;
    const int laneN   = lane & 15;
    const int halfSel = lane >> 4;              // 0: K lo pair, 1: K hi pair

    const int r0 = blockIdx.x * 16;             // 16 query rows per WG
    const int b  = r0 >> 10;
    const int p0 = r0 & 1023;
    const float* fBase = features + (size_t)b * (C_DIM * HW_DIM) + p0;

    // ---- resident A panel via async b32 DMA (per-lane scatter-transpose) --
    // sA[row][k] = features[b, k, p0+row];  64 DMA instructions per wave.
#pragma unroll 8
    for (int it = 0; it < 64; ++it) {
        int id  = tid + 256 * it;               // 0..16383
        int row = id & 15;
        int k   = id >> 4;                      // 0..1023
        async_ld_b32(fBase + (size_t)k * HW_DIM + row, sA + row * ASTR + k);
    }

    float fsq[8];
#pragma unroll
    for (int j = 0; j < 8; ++j) fsq[j] = fsqArr[r0 + j + 8 * halfSel];

    float bd[8];
    int   bi[8];
#pragma unroll
    for (int j = 0; j < 8; ++j) { bd[j] = 3.4e38f; bi[j] = 0; }

    for (int m0 = 0; m0 < M_DIM; m0 += 128) {
        const int colBase = m0 + wave * 16;
        const int myCol   = colBase + laneN;
        const float msq_l = msqArr[myCol];
        v8f acc = {0.f, 0.f, 0.f, 0.f, 0.f, 0.f, 0.f, 0.f};

        // stage a B chunk [128 cols][64 k] into buffer `buf` via async DMA
        auto stageB = [&](int kc, int buf) {
            const int k0 = kc * 64;
            float* dst = sB + buf * (128 * BSTR);
#pragma unroll
            for (int it = 0; it < 8; ++it) {
                int id = tid + 256 * it;        // 0..2047 float4 tiles
                int m  = id >> 4;               // 0..127
                int j4 = id & 15;               // float4 along k
                async_ld_b128(mb + (size_t)(m0 + m) * C_DIM + k0 + j4 * 4,
                              dst + m * BSTR + j4 * 4);
            }
        };

        stageB(0, 0);

        for (int kc = 0; kc < 16; ++kc) {
            if (kc < 15) {
                stageB(kc + 1, (kc + 1) & 1);   // overlap next chunk's DMA
                wait_async_le8();               // chunk kc (and older) landed
            } else {
                wait_async_0();
            }
            __syncthreads();

            const float* bBuf = sB + (kc & 1) * (128 * BSTR);
            const int k0 = kc * 64;
            // ---- 16 WMMA steps over this K chunk ----
#pragma unroll
            for (int kk = 0; kk < 64; kk += 4) {
                const int ke = kk + 2 * halfSel;
                v2f a  = *(const v2f*)(sA + laneN * ASTR + k0 + ke);
                v2f bb = *(const v2f*)(bBuf + (wave * 16 + laneN) * BSTR + ke);
                acc = __builtin_amdgcn_wmma_f32_16x16x4_f32(
                        false, a, false, bb, (short)0, acc, false, false);
            }
            __syncthreads();                    // before buffer is re-filled
        }

        // ---- fused distance + per-row argmin over this wave's 16 cols ----
#pragma unroll
        for (int j = 0; j < 8; ++j) {
            float d = fsq[j] + msq_l - 2.0f * acc[j];
            d = sqrtf(fmaxf(d, 0.0f) + EPS_F);
            int idx = myCol;
#pragma unroll
            for (int off = 1; off < 16; off <<= 1) {
                float od = __shfl_xor(d,   off, 32);
                int   oi = __shfl_xor(idx, off, 32);
                if (od < d || (od == d && oi < idx)) { d = od; idx = oi; }
            }
            if (d < bd[j] || (d == bd[j] && idx < bi[j])) { bd[j] = d; bi[j] = idx; }
        }
    }

    // ---- cross-wave argmin reduction (8 waves share the same 16 rows) ----
    if (laneN == 0) {
#pragma unroll
        for (int j = 0; j < 8; ++j) {
            redD[wave * 16 + j + 8 * halfSel] = bd[j];
            redI[wave * 16 + j + 8 * halfSel] = bi[j];
        }
    }
    __syncthreads();
    if (tid < 16) {
        float d = redD[tid];
        int idx = redI[tid];
        for (int w = 1; w < 8; ++w) {
            float od = redD[w * 16 + tid];
            int   oi = redI[w * 16 + tid];
            if (od < d || (od == d && oi < idx)) { d = od; idx = oi; }
        }
        outDist[r0 + tid] = d;
        outIdx[r0 + tid]  = idx;
    }
}

// ---------------------------------------------------------------------------
// Kernel 3: influence_per_pos[n] = mean_c |f[n,c] - mb[m*,c]| / d0
// ---------------------------------------------------------------------------
__global__ __launch_bounds__(256)
void influence_kernel(const float* __restrict__ features,
                      const float* __restrict__ mb,
                      const float* __restrict__ bestD,
                      const int*   __restrict__ bestI,
                      float* __restrict__ infRaw)
{
    __shared__ float red[256];
    const int n = blockIdx.x;
    const int b = n >> 10, p = n & 1023;
    const int m = bestI[n];
    const float* fb = features + (size_t)b * (C_DIM * HW_DIM) + p;
    const float* mr = mb + (size_t)m * C_DIM;
    float s = 0.0f;
    for (int c = threadIdx.x; c < C_DIM; c += 256)
        s += fabsf(fb[(size_t)c * HW_DIM] - mr[c]);
    red[threadIdx.x] = s;
    __syncthreads();
    for (int st = 128; st > 0; st >>= 1) {
        if (threadIdx.x < st) red[threadIdx.x] += red[threadIdx.x + st];
        __syncthreads();
    }
    if (threadIdx.x == 0) {
        float d0 = bestD[n] + EPS_F;
        infRaw[n] = red[0] / (1024.0f * d0);
    }
}

// ---------------------------------------------------------------------------
// Kernel 4: global min/max of influence_per_pos (8192 values, single WG)
// ---------------------------------------------------------------------------
__global__ __launch_bounds__(256)
void minmax_kernel(const float* __restrict__ infRaw, float* __restrict__ mm)
{
    __shared__ float rmin[256], rmax[256];
    float lo = 3.4e38f, hi = -3.4e38f;
    for (int i = threadIdx.x; i < N_ROWS; i += 256) {
        float v = infRaw[i];
        lo = fminf(lo, v);
        hi = fmaxf(hi, v);
    }
    rmin[threadIdx.x] = lo;
    rmax[threadIdx.x] = hi;
    __syncthreads();
    for (int st = 128; st > 0; st >>= 1) {
        if (threadIdx.x < st) {
            rmin[threadIdx.x] = fminf(rmin[threadIdx.x], rmin[threadIdx.x + st]);
            rmax[threadIdx.x] = fmaxf(rmax[threadIdx.x], rmax[threadIdx.x + st]);
        }
        __syncthreads();
    }
    if (threadIdx.x == 0) { mm[0] = rmin[0]; mm[1] = rmax[0]; }
}

// ---------------------------------------------------------------------------
// Kernel 5: noised output (B,C,H,W) + influence_map + noise_std_map
// ---------------------------------------------------------------------------
__global__ __launch_bounds__(256)
void output_kernel(const float* __restrict__ features,
                   const float* __restrict__ noise,
                   const float* __restrict__ infRaw,
                   const float* __restrict__ mm,
                   float* __restrict__ out)
{
    const int idx = blockIdx.x * 256 + threadIdx.x;   // < 8388608
    const int b   = idx >> 20;                        // / (C*HW)
    const int rem = idx & 1048575;
    const int c   = rem >> 10;
    const int p   = rem & 1023;
    const int n   = (b << 10) | p;

    const float imin  = mm[0];
    const float imax  = mm[1];
    const float denom = imax - imin;
    const float inf   = infRaw[n];
    const float norm  = (denom > EPS_F) ? (inf - imin) / fmaxf(denom, EPS_F) : 0.0f;
    const float stdv  = NOISE_MIN_F + norm * NOISE_RANGE_F;

    out[idx] = features[idx] + noise[(size_t)n * C_DIM + c] * stdv;
    if (c == 0) {
        out[8388608 + n]        = norm;   // influence_map
        out[8388608 + 8192 + n] = stdv;   // noise_std_map
    }
}

// ---------------------------------------------------------------------------
// Launcher
// ---------------------------------------------------------------------------
extern "C" void kernel_launch(void* const* d_in, const int* in_sizes, int n_in,
                              void* d_out, int out_size, void* d_ws, size_t ws_size,
                              hipStream_t stream)
{
    const float* features = (const float*)d_in[0];   // (8,1024,32,32)
    const float* mb       = (const float*)d_in[1];   // (16384,1024)
    const float* noise    = (const float*)d_in[2];   // (8192,1024)
    float* out = (float*)d_out;

    // workspace layout (floats): fsq[8192] | msq[16384] | bestD[8192] |
    // bestI[8192 ints] | infRaw[8192] | mm[2]   (~197 KB total)
    float* ws      = (float*)d_ws;
    float* fsqArr  = ws;
    float* msqArr  = ws + 8192;
    float* bestD   = ws + 24576;
    int*   bestI   = (int*)(ws + 32768);
    float* infRaw  = ws + 40960;
    float* mm      = ws + 49152;

    // 1) row norms: (8192 + 16384) rows, one wave each, 8 waves per block
    norms_kernel<<<(N_ROWS + M_DIM) / 8, 256, 0, stream>>>(features, mb, fsqArr, msqArr);

    // 2) fused WMMA GEMM + argmin: 512 WGs x 256 thr, ~134 KB dynamic LDS
    //    (CDNA5 WGP has 320 KB LDS)
    const size_t ldsBytes = (size_t)(16 * ASTR + 2 * 128 * BSTR + 128) * sizeof(float)
                            + 128 * sizeof(int);
    knn_gemm_kernel<<<N_ROWS / 16, 256, ldsBytes, stream>>>(
        features, mb, fsqArr, msqArr, bestD, bestI);

    // 3) influence per position
    influence_kernel<<<N_ROWS, 256, 0, stream>>>(features, mb, bestD, bestI, infRaw);

    // 4) global min/max
    minmax_kernel<<<1, 256, 0, stream>>>(infRaw, mm);

    // 5) final outputs
    output_kernel<<<8388608 / 256, 256, 0, stream>>>(features, noise, infRaw, mm, out);
}